// ASGLoss_12120397709953
// MI455X (gfx1250) — compile-verified
//
#include <hip/hip_runtime.h>
#include <hip/hip_bf16.h>

#define T_DIM 512
#define B_DIM 32
#define C_DIM 512
#define S_DIM 256
#define NEGV  (-1e30f)

typedef float v2f __attribute__((ext_vector_type(2)));
typedef float v8f __attribute__((ext_vector_type(8)));

// ---------------------------------------------------------------------------
// Pre-pack exp(transition) into the WMMA_F32_16X16X4_F32 A-operand layout.
// A tile (jt, ks) is 16 rows (j) x 4 cols (i=K). Lane l (<16): M=l, K=c;
// lane l (>=16): M=l-16, K=2+c  (c = VGPR index 0/1).
// Linear layout: Apack[((jt*128 + ks)*32 + lane)*2 + c]  -> per-wave loads are
// fully coalesced global_load_b64.
// ---------------------------------------------------------------------------
__global__ void prep_apack_kernel(const float* __restrict__ trans,
                                  float* __restrict__ apack) {
  int tid  = blockIdx.x * blockDim.x + threadIdx.x;   // 0 .. C*C-1
  int c    =  tid        & 1;
  int lane = (tid >> 1)  & 31;
  int ks   = (tid >> 6)  & 127;
  int jt   =  tid >> 13;                               // 0..31
  int j = jt * 16 + (lane & 15);
  int i = ks * 4 + 2 * (lane >> 4) + c;
  apack[tid] = expf(trans[j * C_DIM + i]);
}

// ---------------------------------------------------------------------------
// logsumexp over a full alpha column (one batch) by one wave.
// alphaS has row pitch 17 (bank-conflict-free for this access pattern).
// ---------------------------------------------------------------------------
__device__ __forceinline__ void capture_logZ(const float* alphaS, int col,
                                             int lane, float* dst) {
  float lm = NEGV;
#pragma unroll
  for (int r = 0; r < 16; ++r)
    lm = fmaxf(lm, alphaS[(lane + 32 * r) * 17 + col]);
#pragma unroll
  for (int off = 16; off > 0; off >>= 1)
    lm = fmaxf(lm, __shfl_xor(lm, off, 32));
  float s = 0.0f;
#pragma unroll
  for (int r = 0; r < 16; ++r)
    s += expf(alphaS[(lane + 32 * r) * 17 + col] - lm);
#pragma unroll
  for (int off = 16; off > 0; off >>= 1)
    s += __shfl_xor(s, off, 32);
  if (lane == 0) *dst = lm + logf(s);
}

// ---------------------------------------------------------------------------
// Fully-connected forward DP (logZ).  One workgroup handles 16 batches.
// Per step:  alpha' = m + log( exp(trans) @ exp(alpha - m) ) + x_t
// The matmul is 512x512 @ 512x16 done as 32 WMMA 16x16 tiles (one per wave),
// K=512 -> 128 x V_WMMA_F32_16X16X4_F32 per tile.
//
// LDS layout is manual so that ebuf sits at offset 0: all inner-loop DS
// addresses (< 32 KB) fold into the 16-bit ds_load immediate offset, and the
// K-pair-interleaved ebuf layout makes each lane's two B-operand values
// adjacent -> a single ds_load_b64 per WMMA.
//   ebuf [(i>>1)*32 + 2*b + (i&1)]   : exp(alpha-m), 32 KB, offset 0
//   alpha[j*17 + b]                  : pitch 17 (conflict-free), 34 KB
//   xemis[j*17 + b]                  : emissions x_t, 34 KB
//   mrow [b]                         : per-batch running max
// ---------------------------------------------------------------------------
#define EBUF_F  (C_DIM * 16)
#define ALPHA_F (C_DIM * 17)
#define XEMIS_F (C_DIM * 17)

__global__ __launch_bounds__(1024, 1) void fcc_kernel(
    const float* __restrict__ inputs, const float* __restrict__ apack,
    const int* __restrict__ len_in, float* __restrict__ fcc_out) {
  __shared__ float smem[EBUF_F + ALPHA_F + XEMIS_F + 16];
  float* const ebuf  = smem;                 // LDS offset 0
  float* const alpha = smem + EBUF_F;
  float* const xemis = alpha + ALPHA_F;
  float* const mrow  = xemis + XEMIS_F;

  const int tid    = threadIdx.x;
  const int lane   = tid & 31;
  const int wave   = tid >> 5;          // 0..31
  const int laneLo = lane & 15;
  const int laneHi = lane >> 4;
  const int b0     = blockIdx.x * 16;

  // ---- t = 0: alpha[j][b] = inputs[0][b0+b][j]  (coalesced stage)
  {
    const int bb = wave & 15;
    const float* src = inputs + (0 * B_DIM + b0 + bb) * C_DIM;
#pragma unroll
    for (int rr = 0; rr < 8; ++rr) {
      int j = lane + 32 * ((wave >> 4) * 8 + rr);
      alpha[j * 17 + bb] = src[j];
    }
  }
  __syncthreads();

  const int myLenM1 = len_in[b0 + (wave & 15)] - 1;

  for (int t = 1; t < T_DIM; ++t) {
    // ---- capture logZ for batches whose sequence ended at t-1 (reads alpha)
    if (wave < 16 && myLenM1 == t - 1)
      capture_logZ(alpha, wave, lane, fcc_out + b0 + wave);

    // ---- phase 1: per-batch max over alpha column (waves 0..15)
    if (wave < 16) {
      float lm = NEGV;
#pragma unroll
      for (int r = 0; r < 16; ++r)
        lm = fmaxf(lm, alpha[(lane + 32 * r) * 17 + wave]);
#pragma unroll
      for (int off = 16; off > 0; off >>= 1)
        lm = fmaxf(lm, __shfl_xor(lm, off, 32));
      if (lane == 0) mrow[wave] = lm;
    }
    __syncthreads();

    // ---- phase 2a: e = exp(alpha - m), K-pair interleaved for b64 B loads
    {
      int b  = tid & 15;
      int i0 = tid >> 4;                 // 0..63
      float mb = mrow[b];
#pragma unroll
      for (int r = 0; r < 8; ++r) {
        int i = i0 + 64 * r;
        ebuf[(i >> 1) * 32 + 2 * b + (i & 1)] =
            expf(alpha[i * 17 + b] - mb);
      }
    }
    // ---- phase 2b: stage emissions x_t (coalesced along j)
    {
      const int bb = wave & 15;
      const float* src = inputs + (t * B_DIM + b0 + bb) * C_DIM;
#pragma unroll
      for (int rr = 0; rr < 8; ++rr) {
        int j = lane + 32 * ((wave >> 4) * 8 + rr);
        xemis[j * 17 + bb] = src[j];
      }
    }
    __syncthreads();

    // ---- phase 3: P = exp(trans) @ e  via WMMA f32 16x16x4 (tile jt = wave)
    v8f acc = {0.f, 0.f, 0.f, 0.f, 0.f, 0.f, 0.f, 0.f};
    const float* ap = apack + (wave * 128) * 64 + lane * 2;
    const int bb0 = laneHi * 32 + 2 * laneLo;   // b64-aligned B operand base
    for (int ks = 0; ks < 128; ++ks) {
      v2f a  = *(const v2f*)(ap + ks * 64);
      v2f bv = *(const v2f*)(ebuf + ks * 64 + bb0);  // rows k0,k0+1 for col b
      acc = __builtin_amdgcn_wmma_f32_16x16x4_f32(
          false, a, false, bv, (short)0, acc, false, false);
    }

    // ---- phase 4: alpha' = m_b + log(P) + x_t   (D tile: row v+8*laneHi)
    {
      float mb = mrow[laneLo];
      int j0 = wave * 16 + 8 * laneHi;
#pragma unroll
      for (int v = 0; v < 8; ++v) {
        int j = j0 + v;
        alpha[j * 17 + laneLo] = mb + logf(acc[v]) + xemis[j * 17 + laneLo];
      }
    }
    __syncthreads();
  }

  // ---- final capture (t = T-1)
  if (wave < 16 && myLenM1 == T_DIM - 1)
    capture_logZ(alpha, wave, lane, fcc_out + b0 + wave);
}

// ---------------------------------------------------------------------------
// Forced-alignment DP along target path. One block per batch, thread = state s.
// Double-buffered LDS alpha -> one barrier per step.
// ---------------------------------------------------------------------------
__global__ __launch_bounds__(S_DIM, 1) void fac_kernel(
    const float* __restrict__ inputs, const float* __restrict__ trans,
    const int* __restrict__ targets, const int* __restrict__ len_in,
    const int* __restrict__ len_tgt, float* __restrict__ fac_out) {
  __shared__ float al[2][S_DIM];

  const int b = blockIdx.x;
  const int s = threadIdx.x;

  const int y     = targets[b * S_DIM + s];
  const int yprev = (s > 0) ? targets[b * S_DIM + s - 1] : y;
  const float self_t = trans[y * C_DIM + y];
  const float adv_t  = trans[y * C_DIM + yprev];
  const int Lin = len_in[b];
  const int Ltg = len_tgt[b];

  float a0 = (s == 0) ? inputs[(0 * B_DIM + b) * C_DIM + y] : NEGV;
  al[0][s] = a0;
  if (Lin == 1 && s == Ltg - 1) fac_out[b] = a0;
  __syncthreads();

  for (int t = 1; t < T_DIM; ++t) {
    const float* cur = al[(t - 1) & 1];
    float* nxt = al[t & 1];
    float e = inputs[(t * B_DIM + b) * C_DIM + y];
    float stay = cur[s] + self_t;
    float move = (s > 0) ? (cur[s - 1] + adv_t) : NEGV;
    float mx = fmaxf(stay, move);
    float nv = mx + log1pf(expf(fminf(stay, move) - mx)) + e;
    nxt[s] = nv;
    if (t == Lin - 1 && s == Ltg - 1) fac_out[b] = nv;
    __syncthreads();
  }
}

// ---------------------------------------------------------------------------
// out = mean_b(fcc[b] - fac[b])
// ---------------------------------------------------------------------------
__global__ void reduce_kernel(const float* __restrict__ fcc,
                              const float* __restrict__ fac,
                              float* __restrict__ out) {
  int l = threadIdx.x;   // 32 threads = 1 wave
  float v = fcc[l] - fac[l];
#pragma unroll
  for (int off = 16; off > 0; off >>= 1) v += __shfl_xor(v, off, 32);
  if (l == 0) out[0] = v * (1.0f / (float)B_DIM);
}

// ---------------------------------------------------------------------------
extern "C" void kernel_launch(void* const* d_in, const int* in_sizes, int n_in,
                              void* d_out, int out_size, void* d_ws,
                              size_t ws_size, hipStream_t stream) {
  (void)in_sizes; (void)n_in; (void)out_size; (void)ws_size;
  const float* trans   = (const float*)d_in[0];  // [C,C]
  const float* inputs  = (const float*)d_in[1];  // [T,B,C]
  const int*   targets = (const int*)d_in[2];    // [B,S]
  const int*   len_in  = (const int*)d_in[3];    // [B]
  const int*   len_tgt = (const int*)d_in[4];    // [B]
  float*       out     = (float*)d_out;

  float* apack = (float*)d_ws;                   // C*C floats = 1 MB
  float* fcc   = apack + C_DIM * C_DIM;          // B floats
  float* fac   = fcc + B_DIM;                    // B floats

  prep_apack_kernel<<<(C_DIM * C_DIM) / 256, 256, 0, stream>>>(trans, apack);
  fac_kernel<<<B_DIM, S_DIM, 0, stream>>>(inputs, trans, targets, len_in,
                                          len_tgt, fac);
  fcc_kernel<<<B_DIM / 16, 1024, 0, stream>>>(inputs, apack, len_in, fcc);
  reduce_kernel<<<1, 32, 0, stream>>>(fcc, fac, out);
}